// DescriptorMatcher_8383776161895
// MI455X (gfx1250) — compile-verified
//
#include <hip/hip_runtime.h>
#include <hip/hip_bf16.h>

typedef float v2f __attribute__((ext_vector_type(2)));
typedef float v8f __attribute__((ext_vector_type(8)));

#define DIMD   128
#define WAVES  8
#define TH_SNN 0.8f
#define BIGF   3.402823466e+38f

// ---------------------------------------------------------------------------
// Kernel 1: squared L2 norm of every row of d1 (n1 rows) and d2 (n2 rows).
// sq[0..n1-1] = ||d1_i||^2 ; sq[n1..n1+n2-1] = ||d2_j||^2
// ---------------------------------------------------------------------------
__global__ __launch_bounds__(256) void rowsq_kernel(const float* __restrict__ d1,
                                                    const float* __restrict__ d2,
                                                    float* __restrict__ sq,
                                                    int n1, int n2) {
    int tid = blockIdx.x * blockDim.x + threadIdx.x;
    if (tid >= n1 + n2) return;
    const float* row = (tid < n1) ? (d1 + (size_t)tid * DIMD)
                                  : (d2 + (size_t)(tid - n1) * DIMD);
    const float4* r4 = (const float4*)row;
    float s = 0.0f;
#pragma unroll
    for (int k = 0; k < DIMD / 4; ++k) {
        float4 v = r4[k];
        s += v.x * v.x + v.y * v.y + v.z * v.z + v.w * v.w;
    }
    sq[tid] = s;
}

// ---------------------------------------------------------------------------
// Kernel 2: fused GEMM (f32 WMMA 16x16x4) + per-row top-2 + snn ratio test.
// One block = 16 rows of d1; 8 waves sweep all column tiles of d2.
// v[i][j] = ||b_j||^2 - 2*dot(a_i,b_j)   (row-constant ||a_i||^2 added at end)
// ---------------------------------------------------------------------------
__global__ __launch_bounds__(256) void snn_match_kernel(const float* __restrict__ d1,
                                                        const float* __restrict__ d2,
                                                        const float* __restrict__ sq,
                                                        float* __restrict__ out,
                                                        int n1, int n2) {
    __shared__ float s_v1[16][WAVES * 16];
    __shared__ float s_v2[16][WAVES * 16];
    __shared__ int   s_i1[16][WAVES * 16];

    const int rowBase = blockIdx.x * 16;
    const int lane    = threadIdx.x & 31;
    const int wave    = threadIdx.x >> 5;
    const int hlf     = lane >> 4;   // 0: lanes 0-15, 1: lanes 16-31
    const int l16     = lane & 15;

    // A fragment source: lane l16 owns row M=l16; lane half selects K pair.
    const float* aRow = d1 + (size_t)(rowBase + l16) * DIMD;

    float v1[8], v2[8];
    int   i1[8];
#pragma unroll
    for (int r = 0; r < 8; ++r) { v1[r] = BIGF; v2[r] = BIGF; i1[r] = -1; }

    const int nTiles = n2 >> 4;
    for (int ct = wave; ct < nTiles; ct += WAVES) {
        const int colBase = ct * 16;
        // B = d2^T, so B[k][n] = d2[n][k]: symmetric float2 row loads.
        const float* bRow = d2 + (size_t)(colBase + l16) * DIMD;

        v8f acc0 = {}; v8f acc1 = {};   // two chains to break WMMA RAW latency
#pragma unroll
        for (int k = 0; k < DIMD; k += 8) {
            v2f a0 = *(const v2f*)(aRow + k + 2 * hlf);
            v2f b0 = *(const v2f*)(bRow + k + 2 * hlf);
            acc0 = __builtin_amdgcn_wmma_f32_16x16x4_f32(
                false, a0, false, b0, (short)0, acc0, false, false);
            v2f a1 = *(const v2f*)(aRow + k + 4 + 2 * hlf);
            v2f b1 = *(const v2f*)(bRow + k + 4 + 2 * hlf);
            acc1 = __builtin_amdgcn_wmma_f32_16x16x4_f32(
                false, a1, false, b1, (short)0, acc1, false, false);
        }
        v8f dot;
#pragma unroll
        for (int r = 0; r < 8; ++r) dot[r] = acc0[r] + acc1[r];

        // Each lane holds column n = l16 of this tile for 8 row values.
        const int   j  = colBase + l16;
        const float c2 = sq[n1 + j];
#pragma unroll
        for (int r = 0; r < 8; ++r) {
            float v = c2 - 2.0f * dot[r];
            if (v < v1[r])      { v2[r] = v1[r]; v1[r] = v; i1[r] = j; }
            else if (v < v2[r]) { v2[r] = v; }
        }
    }

    // Spill per-lane top-2 candidates: C/D layout -> row = r + hlf*8.
    const int col = wave * 16 + l16;
#pragma unroll
    for (int r = 0; r < 8; ++r) {
        const int row = r + hlf * 8;
        s_v1[row][col] = v1[r];
        s_v2[row][col] = v2[r];
        s_i1[row][col] = i1[r];
    }
    __syncthreads();

    // Final merge + ratio test: 16 threads, one per row.
    if (threadIdx.x < 16) {
        const int row = threadIdx.x;
        float bv1 = BIGF, bv2 = BIGF;
        int   bi  = -1;
        for (int c = 0; c < WAVES * 16; ++c) {
            const float cv1 = s_v1[row][c];
            const float cv2 = s_v2[row][c];
            const int   ci  = s_i1[row][c];
            if (cv1 < bv1) { bv2 = fminf(bv1, cv2); bv1 = cv1; bi = ci; }
            else           { bv2 = fminf(bv2, cv1); }
        }
        const int   gr    = rowBase + row;
        const float a2    = sq[gr];
        const float dist1 = sqrtf(fmaxf(a2 + bv1, 0.0f));
        const float dist2 = sqrtf(fmaxf(a2 + bv2, 0.0f));
        const float ratio = dist1 / dist2;
        const bool  m     = (ratio <= TH_SNN);

        // Outputs concatenated flat: match_dists (n1), matches_idxs (n1*2), mask (n1)
        out[gr]                   = m ? ratio : 0.0f;
        out[n1 + 2 * gr]          = m ? (float)gr : -1.0f;
        out[n1 + 2 * gr + 1]      = m ? (float)bi : -1.0f;
        out[n1 + 2 * n1 + gr]     = m ? 1.0f : 0.0f;
    }
}

extern "C" void kernel_launch(void* const* d_in, const int* in_sizes, int n_in,
                              void* d_out, int out_size, void* d_ws, size_t ws_size,
                              hipStream_t stream) {
    const float* d1 = (const float*)d_in[0];
    const float* d2 = (const float*)d_in[1];
    float* sq  = (float*)d_ws;     // n1 + n2 floats of row norms
    float* out = (float*)d_out;

    const int n1 = in_sizes[0] / DIMD;   // 8192
    const int n2 = in_sizes[1] / DIMD;   // 8192

    rowsq_kernel<<<(n1 + n2 + 255) / 256, 256, 0, stream>>>(d1, d2, sq, n1, n2);
    snn_match_kernel<<<n1 / 16, 256, 0, stream>>>(d1, d2, sq, out, n1, n2);
}